// CompanionSSM_58188216926763
// MI455X (gfx1250) — compile-verified
//
#include <hip/hip_runtime.h>

// Problem sizes (fixed by reference): B=32, K=64, L=2048, order=8, d=512.
#define KCH 64
#define DDIM 512
#define LSEQ 2048
#define BSZ 32
#define EPSV 1e-8f
#define MIN_NORMV 1e-4f

typedef __attribute__((ext_vector_type(2))) float v2f;
typedef __attribute__((ext_vector_type(8))) float v8f;

// CDNA5 fp32 matrix op: D(16x16,f32) = A(16x4,f32) * B(4x16,f32) + C
__device__ __forceinline__ v8f wmma4(v2f a, v2f b, v8f c) {
  return __builtin_amdgcn_wmma_f32_16x16x4_f32(false, a, false, b, (short)0, c,
                                               false, false);
}

__device__ __forceinline__ void wait_async0() {
#if __has_builtin(__builtin_amdgcn_s_wait_asynccnt)
  __builtin_amdgcn_s_wait_asynccnt(0);
#else
  asm volatile("s_wait_asynccnt 0" ::: "memory");
#endif
}

// ---------------------------------------------------------------------------
// 1) coeff normalization: xn = sum|coeff|; coeffN = coeff/(xn+eps) if |xn|>MIN
__global__ void k_prep_coeff(const float* __restrict__ coeff,
                             float* __restrict__ coeffN) {
  __shared__ float red[256];
  float s = 0.f;
  for (int i = threadIdx.x; i < KCH * DDIM; i += 256) s += fabsf(coeff[i]);
  red[threadIdx.x] = s;
  __syncthreads();
  for (int off = 128; off > 0; off >>= 1) {
    if ((int)threadIdx.x < off) red[threadIdx.x] += red[threadIdx.x + off];
    __syncthreads();
  }
  float xn = red[0];
  float inv = (fabsf(xn) > MIN_NORMV) ? 1.0f / (xn + EPSV) : 1.0f;
  for (int i = threadIdx.x; i < KCH * DDIM; i += 256) coeffN[i] = coeff[i] * inv;
}

// ---------------------------------------------------------------------------
// 2) materialize companion matrix A (512x512)
__global__ void k_build_A(const float* __restrict__ coeffN,
                          float* __restrict__ A) {
  int idx = blockIdx.x * 256 + threadIdx.x;
  if (idx >= DDIM * DDIM) return;
  int i = idx >> 9, j = idx & (DDIM - 1);
  float v = (i < KCH) ? coeffN[i * DDIM + j] : ((j == i - KCH) ? 1.0f : 0.0f);
  A[idx] = v;
}

// ---------------------------------------------------------------------------
// 3) Dst = S @ S  (512x512x512), WMMA f32.
//    ldsA row-major (stride 68), ldsB column-major (stride 68) so that both
//    fragments are contiguous 8B-aligned lane pairs -> single ds_load_b64.
__global__ __launch_bounds__(512) void k_gemm512(const float* __restrict__ S,
                                                 float* __restrict__ Dst) {
  __shared__ float ldsA[64 * 68];
  __shared__ float ldsB[64 * 68];
  const int tid = threadIdx.x;
  const int w = tid >> 5, lane = tid & 31, lo = lane & 15, hi = lane >> 4;
  const int bm = (blockIdx.x >> 3) * 64, bn = (blockIdx.x & 7) * 64;
  const int wm = (w >> 2) * 16, wn = (w & 3) * 16;
  v8f acc = {};
  for (int KK = 0; KK < 512; KK += 64) {
    for (int i = tid; i < 64 * 64; i += 512) {
      int r = i >> 6, c = i & 63;
      ldsA[r * 68 + c] = S[(bm + r) * 512 + KK + c];   // A[m][kk] row-major
      ldsB[c * 68 + r] = S[(KK + r) * 512 + bn + c];   // B[n][kk] col-major
    }
    __syncthreads();
    for (int kk = 0; kk < 16; ++kk) {
      int kg = kk * 4 + 2 * hi;
      v2f a = *(const v2f*)&ldsA[(wm + lo) * 68 + kg];
      v2f b = *(const v2f*)&ldsB[(wn + lo) * 68 + kg];
      acc = wmma4(a, b, acc);
    }
    __syncthreads();
  }
  for (int rr = 0; rr < 8; ++rr) {
    int row = bm + wm + 8 * hi + rr;
    Dst[row * 512 + bn + wn + lo] = acc[rr];
  }
}

// ---------------------------------------------------------------------------
// 4) init: V0 = B_mat (d x K), U0 = C_mat^T (d x K)
__global__ void k_init_UV(const float* __restrict__ Bmat,
                          const float* __restrict__ Cmat,
                          float* __restrict__ Vws, float* __restrict__ Uws) {
  int i = blockIdx.x * 256 + threadIdx.x;  // 32768
  if (i >= DDIM * KCH) return;
  int d = i >> 6, k = i & 63;
  Vws[i] = Bmat[i];
  Uws[i] = Cmat[k * DDIM + d];
}

// ---------------------------------------------------------------------------
// 5) doubling step: for q in [0,2^j): buf[q + 2^j] = op(M) @ buf[q]
//    op = M (V chain) or M^T (U chain). Each 512x512 @ 512x64, WMMA f32.
//    B operand staged column-major (stride 516): per-lane b64, banks 4*lane.
//    Each wave owns a 16-row strip x all 64 cols: A-frag loaded once, fed to
//    4 independent accumulators (4 back-to-back WMMAs per fragment).
template <bool TRANSP>
__global__ __launch_bounds__(256) void k_step(const float* __restrict__ M,
                                              float* __restrict__ buf, int j) {
  extern __shared__ float sIn[];  // [k][d] col-major, stride 516 (132096 B)
  const int n = 1 << j;
  const int q = blockIdx.x >> 2;
  const int quarter = blockIdx.x & 3;
  const int tid = threadIdx.x;
  const float* IN = buf + (size_t)q * (DDIM * KCH);
  float* OUT = buf + (size_t)(q + n) * (DDIM * KCH);
  for (int i = tid; i < DDIM * KCH; i += 256) {
    int d = i >> 6, k = i & 63;
    sIn[k * 516 + d] = IN[i];
  }
  __syncthreads();
  const int w = tid >> 5, lane = tid & 31, lo = lane & 15, hi = lane >> 4;
  const int tm = (quarter * 8 + w) * 16;  // 16-row strip in [0,512)
  v8f acc0 = {}, acc1 = {}, acc2 = {}, acc3 = {};
  for (int kk = 0; kk < 128; ++kk) {
    int kg = kk * 4 + 2 * hi;
    v2f a;
    if (TRANSP) {  // A-frag[m][c] = M[kg+c][tm+m] (coalesced over lanes)
      a.x = M[kg * 512 + tm + lo];
      a.y = M[(kg + 1) * 512 + tm + lo];
    } else {       // A-frag[m][c] = M[tm+m][kg+c] (8B-aligned per-lane b64)
      a = *(const v2f*)&M[(tm + lo) * 512 + kg];
    }
    v2f b0 = *(const v2f*)&sIn[(0 * 16 + lo) * 516 + kg];
    v2f b1 = *(const v2f*)&sIn[(1 * 16 + lo) * 516 + kg];
    v2f b2 = *(const v2f*)&sIn[(2 * 16 + lo) * 516 + kg];
    v2f b3 = *(const v2f*)&sIn[(3 * 16 + lo) * 516 + kg];
    acc0 = wmma4(a, b0, acc0);
    acc1 = wmma4(a, b1, acc1);
    acc2 = wmma4(a, b2, acc2);
    acc3 = wmma4(a, b3, acc3);
  }
  for (int rr = 0; rr < 8; ++rr) {
    int row = tm + 8 * hi + rr;
    OUT[row * 64 + 0 + lo] = acc0[rr];
    OUT[row * 64 + 16 + lo] = acc1[rr];
    OUT[row * 64 + 32 + lo] = acc2[rr];
    OUT[row * 64 + 48 + lo] = acc3[rr];
  }
}

// ---------------------------------------------------------------------------
// 6) kern[k, 32q + r] = sum_d U[q][d][k] * V[r][d][k]   (diag contraction)
__global__ void k_dot(const float* __restrict__ U, const float* __restrict__ V,
                      float* __restrict__ kern) {
  int tid = blockIdx.x * 256 + threadIdx.x;  // 131072
  if (tid >= KCH * LSEQ) return;
  int k = tid & 63;
  int t = tid >> 6;
  int q = t >> 5, r = t & 31;
  const float* u = U + (size_t)q * (DDIM * KCH) + k;
  const float* v = V + (size_t)r * (DDIM * KCH) + k;
  float acc = 0.f;
  for (int d = 0; d < DDIM; ++d) acc += u[d * 64] * v[d * 64];
  kern[k * LSEQ + t] = acc;
}

// ---------------------------------------------------------------------------
// 7) causal conv: y[b,k,l] = sum_{t<=l} u[b,k,t] * kern[k,l-t]
//    Per channel k: u (32x2048) DMA'd into LDS via CDNA5 async-to-LDS B128
//    (ASYNCcnt path; row stride 2052 words = 8208 B keeps every 16B chunk
//    16B-aligned and lanes on distinct banks). Tiled Toeplitz WMMA:
//    D[batch16 x l16] += Afrag(u) x Bfrag(kern gather from zero-padded LDS).
__global__ __launch_bounds__(1024) void k_conv(const float* __restrict__ u,
                                               const float* __restrict__ kern,
                                               float* __restrict__ y) {
  extern __shared__ float smem[];
  float* sU = smem;                // 32 * 2052
  float* sK = smem + 32 * 2052;    // 16 zeros + 2048 kernel taps
  const int k = blockIdx.x;
  const int tid = threadIdx.x;

  // Async DMA: 16384 chunks of 16 B, global -> LDS, tracked by ASYNCcnt.
  const unsigned ldsU = (unsigned)(unsigned long long)(uintptr_t)sU;
  const unsigned long long ubase = (unsigned long long)(uintptr_t)u;
  for (int ci = tid; ci < BSZ * (LSEQ / 4); ci += 1024) {
    int b = ci >> 9, c4 = ci & 511;                       // 512 chunks/row
    unsigned lAddr = ldsU + (unsigned)((b * 2052 + c4 * 4) * 4);
    unsigned gOff = (unsigned)((((b * KCH + k) * LSEQ) + c4 * 4) * 4);
    asm volatile("global_load_async_to_lds_b128 %0, %1, %2"
                 :: "v"(lAddr), "v"(gOff), "s"(ubase)
                 : "memory");
  }
  for (int i = tid; i < 2064; i += 1024)
    sK[i] = (i < 16) ? 0.f : kern[k * LSEQ + (i - 16)];
  wait_async0();
  __syncthreads();

  const int w = tid >> 5, lane = tid & 31, lo = lane & 15, hi = lane >> 4;
  for (int task = w; task < 256; task += 32) {  // 128 l-tiles x 2 batch halves
    int l0 = (task >> 1) * 16;
    int bbase = (task & 1) * 16;
    v8f acc = {};
    int nCh = (l0 >> 2) + 4;  // triangular: only t0 <= l0+12 contributes
    for (int c = 0; c < nCh; ++c) {
      int t0 = c * 4 + 2 * hi;
      v2f a = *(const v2f*)&sU[(bbase + lo) * 2052 + t0];  // 8B-aligned b64
      v2f b;
      b.x = sK[16 + l0 + lo - t0];     // kern[l0+n - t0 - c], pad -> 0
      b.y = sK[16 + l0 + lo - t0 - 1];
      acc = wmma4(a, b, acc);
    }
    for (int rr = 0; rr < 8; ++rr) {
      int b = bbase + 8 * hi + rr;
      y[((size_t)b * KCH + k) * LSEQ + l0 + lo] = acc[rr];  // coalesced in l
    }
  }
}

// ---------------------------------------------------------------------------
extern "C" void kernel_launch(void* const* d_in, const int* in_sizes, int n_in,
                              void* d_out, int out_size, void* d_ws,
                              size_t ws_size, hipStream_t stream) {
  const float* u     = (const float*)d_in[0];  // (32,64,2048)
  const float* coeff = (const float*)d_in[1];  // (1, 32768)
  const float* Bmat  = (const float*)d_in[2];  // (1, 512, 64)
  const float* Cmat  = (const float*)d_in[3];  // (1, 64, 512)
  float* y = (float*)d_out;
  float* ws = (float*)d_ws;

  // workspace layout (floats), total ~24.8 MB
  float* coeffN = ws;                              // 32768
  float* P      = coeffN + 32768;                  // 11 * 262144  (A^(2^i))
  float* Vws    = P + 11 * 262144;                 // 32 * 32768   (A^r b)
  float* Uws    = Vws + 32 * 32768;                // 64 * 32768   ((A^32T)^q c)
  float* kern   = Uws + 64 * 32768;                // 131072

  k_prep_coeff<<<1, 256, 0, stream>>>(coeff, coeffN);
  k_build_A<<<(DDIM * DDIM) / 256, 256, 0, stream>>>(coeffN, P);  // P0 = A
  for (int i = 0; i < 10; ++i)  // P[i+1] = P[i]^2  -> A^2 .. A^1024
    k_gemm512<<<64, 512, 0, stream>>>(P + i * 262144, P + (i + 1) * 262144);
  k_init_UV<<<(DDIM * KCH) / 256, 256, 0, stream>>>(Bmat, Cmat, Vws, Uws);

  size_t stepLds = 64 * 516 * sizeof(float);
  for (int j = 0; j < 5; ++j)   // V_{r+2^j} = A^{2^j} V_r        (r -> 0..31)
    k_step<false><<<(1 << j) * 4, 256, stepLds, stream>>>(P + j * 262144, Vws, j);
  for (int j = 0; j < 6; ++j)   // U_{q+2^j} = (A^{32*2^j})^T U_q (q -> 0..63)
    k_step<true><<<(1 << j) * 4, 256, stepLds, stream>>>(P + (5 + j) * 262144, Uws, j);

  k_dot<<<(KCH * LSEQ) / 256, 256, 0, stream>>>(Uws, Vws, kern);

  size_t convLds = (32 * 2052 + 2064) * sizeof(float);
  k_conv<<<KCH, 1024, convLds, stream>>>(u, kern, y);
}